// SphericalSpectralTimeConv_38405597561178
// MI455X (gfx1250) — compile-verified
//
#include <hip/hip_runtime.h>
#include <hip/hip_bf16.h>

#define PI_F 3.14159265358979323846f

typedef float v2f __attribute__((ext_vector_type(2)));
typedef float v8f __attribute__((ext_vector_type(8)));

// Problem constants (from reference setup_inputs)
// L_in = L_out = 128, nphi = 255, Lf = 64, Mf = 127, Cin = Cout = 64, T = 256
#define NT    128   // theta samples (== L_in)
#define NPHI  255
#define LF    64
#define MF    127   // 2*Lf - 1
#define NC    64
#define C2    128   // re|im packed channels
#define NTEMB 256

// ---------------------------------------------------------------------------
// Kernel 1: quadrature weights q[128] + twiddle tables tw[255]
// ---------------------------------------------------------------------------
__global__ void k_precompute(float* __restrict__ twc, float* __restrict__ tws,
                             float* __restrict__ q) {
    int tid = threadIdx.x;
    for (int k = tid; k < NPHI; k += 256) {
        float ang = 2.f * PI_F * (float)k / (float)NPHI;
        twc[k] = cosf(ang);
        tws[k] = sinf(ang);
    }
    if (tid < NT) {
        float sr  = 0.f;   // wr[tid]
        float sr2 = 0.f;   // wr[254 - tid]
        int k2 = (NPHI - 1) - tid;
        for (int j = 0; j < NPHI; ++j) {
            int midx = (j + 127) % NPHI;   // ifftshift
            int m = midx - 127;
            float wre, wim;
            if (m == 1)              { wre = 0.f;  wim =  PI_F * 0.5f; }
            else if (m == -1)        { wre = 0.f;  wim = -PI_F * 0.5f; }
            else if ((m & 1) == 0)   { wre = 2.f / (1.f - (float)(m * m)); wim = 0.f; }
            else                     { wre = 0.f;  wim = 0.f; }
            float ang = -(float)m * PI_F / (float)NPHI;
            float ca = cosf(ang), sa = sinf(ang);
            float wr_ = wre * ca - wim * sa;
            float wi_ = wre * sa + wim * ca;
            float a1 = 2.f * PI_F * (float)((j * tid) % NPHI) / (float)NPHI;
            sr  = fmaf(wr_, cosf(a1), fmaf(wi_, sinf(a1), sr));
            float a2 = 2.f * PI_F * (float)((j * k2) % NPHI) / (float)NPHI;
            sr2 = fmaf(wr_, cosf(a2), fmaf(wi_, sinf(a2), sr2));
        }
        float wr_t  = sr  / (float)NPHI;
        float wr_t2 = sr2 / (float)NPHI;
        float qq = wr_t + ((tid < NT - 1) ? wr_t2 : 0.f);
        q[tid] = qq * 2.f * PI_F / (float)NPHI;
    }
}

// ---------------------------------------------------------------------------
// Kernel 2: normalized associated Legendre table P[l, m, t], 0<=m,l<64, t<128
//           layout P[(l*64 + m)*128 + t]; zeros for l < m.
// ---------------------------------------------------------------------------
__global__ void k_legendre(float* __restrict__ P) {
    int m = blockIdx.x;       // 0..63
    int t = threadIdx.x;      // 0..127
    float theta = PI_F * (float)(2 * t + 1) / (float)NPHI;
    float ct = cosf(theta), st = sinf(theta);
    for (int l = 0; l < m; ++l) P[((size_t)l * LF + m) * NT + t] = 0.f;
    float pmm = 0.28209479177387814f;     // 1/sqrt(4*pi)
    for (int k = 1; k <= m; ++k)
        pmm *= -sqrtf((2.f * k + 1.f) / (2.f * k)) * st;
    P[((size_t)m * LF + m) * NT + t] = pmm;
    if (m + 1 < LF) {
        float pl1 = sqrtf(2.f * m + 3.f) * ct * pmm;
        P[((size_t)(m + 1) * LF + m) * NT + t] = pl1;
        float pl2 = pmm;
        for (int l = m + 2; l < LF; ++l) {
            float ll = (float)l;
            float mm = (float)(m * m);
            float a = sqrtf((4.f * ll * ll - 1.f) / (ll * ll - mm));
            float b = sqrtf((2.f * ll + 1.f) * ((ll - 1.f) * (ll - 1.f) - mm) /
                            ((2.f * ll - 3.f) * (ll * ll - mm)));
            float pl = a * ct * pl1 - b * pl2;
            P[((size_t)l * LF + m) * NT + t] = pl;
            pl2 = pl1; pl1 = pl;
        }
    }
}

// ---------------------------------------------------------------------------
// Kernel 3: forward phi-DFT with quad weight + neg-m sign folded in.
//   FW[mi][t][c2] = q[t]*sign(m) * DFT_p(x[t,p,c]),  mi = m+63, c2 = re|im
// One block per theta row t; x[t] staged in LDS (65,280 B).
// ---------------------------------------------------------------------------
__global__ void k_dft_fwd(const float* __restrict__ x,
                          const float* __restrict__ twc,
                          const float* __restrict__ tws,
                          const float* __restrict__ q,
                          float* __restrict__ FW) {
    __shared__ float xs[NPHI * NC];
    int t = blockIdx.x;
    int tid = threadIdx.x;
    const float* xrow = x + (size_t)t * NPHI * NC;
    for (int i = tid; i < NPHI * NC; i += 256) xs[i] = xrow[i];
    __syncthreads();
    float qt = q[t];
    for (int idx = tid; idx < MF * NC; idx += 256) {
        int mi = idx >> 6;          // 0..126
        int c  = idx & 63;
        int ms = mi - 63;           // -63..63
        int mmod = (ms < 0) ? ms + NPHI : ms;
        float ar = 0.f, ai = 0.f;
        int k = 0;
        for (int p = 0; p < NPHI; ++p) {
            float v = xs[p * NC + c];
            ar = fmaf(v,  twc[k], ar);
            ai = fmaf(v, -tws[k], ai);       // e^{-i 2pi m p / 255}
            k += mmod; if (k >= NPHI) k -= NPHI;
        }
        float sign = (ms < 0 && (ms & 1)) ? -1.f : 1.f;   // (-1)^|m| for m<0
        float s = qt * sign;
        float* dst = FW + ((size_t)mi * NT + t) * C2;
        dst[c]      = ar * s;
        dst[NC + c] = ai * s;
    }
}

// ---------------------------------------------------------------------------
// Kernel 4: forward Legendre contraction via f32 WMMA, one block per m.
//   FLM[m][l][c2] = sum_t P[l,|m|,t] * FW[m][t][c2]     (64 x 128 x 128 GEMM)
// ---------------------------------------------------------------------------
__global__ void k_gemm_fwd(const float* __restrict__ P,
                           const float* __restrict__ FW,
                           float* __restrict__ FLM) {
    int m = blockIdx.x;                       // 0..126
    int mabs = (m < 63) ? (63 - m) : (m - 63);
    int lane  = threadIdx.x & 31;
    int wave  = threadIdx.x >> 5;             // 0..7 -> N tile
    int lhalf = lane >> 4;                    // 0 or 1
    int l15   = lane & 15;
    int n0    = wave * 16;
    const float* Bm = FW  + (size_t)m * NT * C2;   // [t][c2]
    float*       Cm = FLM + (size_t)m * LF * C2;   // [l][c2]
    for (int r = 0; r < 4; ++r) {
        int l0 = r * 16;
        v8f acc = {};
        for (int k0 = 0; k0 < NT; k0 += 4) {
            int ka = k0 + lhalf * 2;
            int arow = l0 + l15;
            v2f a, b;
            a.x = P[((size_t)arow * LF + mabs) * NT + ka];
            a.y = P[((size_t)arow * LF + mabs) * NT + ka + 1];
            b.x = Bm[(size_t)ka * C2 + n0 + l15];
            b.y = Bm[(size_t)(ka + 1) * C2 + n0 + l15];
            acc = __builtin_amdgcn_wmma_f32_16x16x4_f32(
                false, a, false, b, (short)0, acc, false, false);
        }
#pragma unroll
        for (int rv = 0; rv < 8; ++rv) {
            int lrow = l0 + rv + lhalf * 8;
            Cm[(size_t)lrow * C2 + n0 + l15] = acc[rv];
        }
    }
}

// ---------------------------------------------------------------------------
// Kernel 5: t_c[m][l] = (t_emb . A_real[l,m,:], t_emb . A_imag[l,m,:])
// ---------------------------------------------------------------------------
__global__ void k_modulate(const float* __restrict__ temb,
                           const float* __restrict__ Ar,
                           const float* __restrict__ Ai,
                           float* __restrict__ TC) {
    int m = blockIdx.x, l = threadIdx.x;
    const float* ar = Ar + ((size_t)l * MF + m) * NTEMB;
    const float* ai = Ai + ((size_t)l * MF + m) * NTEMB;
    float re = 0.f, im = 0.f;
    for (int t = 0; t < NTEMB; ++t) {
        float e = temb[t];
        re = fmaf(e, ar[t], re);
        im = fmaf(e, ai[t], im);
    }
    TC[((size_t)m * LF + l) * 2 + 0] = re;
    TC[((size_t)m * LF + l) * 2 + 1] = im;
}

// ---------------------------------------------------------------------------
// Kernel 6: Rt[l][o][i] = sum_m t_c[l,m] * R[l,m,o,i]   (complex)
// Streams 266 MB of R exactly once, coalesced; this is the roofline term.
// grid (16, 64): y = l, x = 256-wide (o,i) chunk.
// ---------------------------------------------------------------------------
__global__ void k_mixR(const float* __restrict__ TC,
                       const float* __restrict__ Rr,
                       const float* __restrict__ Ri,
                       float* __restrict__ RTr,
                       float* __restrict__ RTi) {
    __shared__ float tcs[MF * 2];
    int l   = blockIdx.y;
    int tid = threadIdx.x;
    if (tid < MF) {
        tcs[2 * tid]     = TC[((size_t)tid * LF + l) * 2 + 0];
        tcs[2 * tid + 1] = TC[((size_t)tid * LF + l) * 2 + 1];
    }
    __syncthreads();
    int oi = blockIdx.x * 256 + tid;                   // 0..4095
    const float* rr = Rr + (size_t)l * MF * 4096 + oi;
    const float* ri = Ri + (size_t)l * MF * 4096 + oi;
    float ar = 0.f, ai = 0.f;
    for (int m = 0; m < MF; ++m) {
        if (m + 8 < MF) {
            __builtin_prefetch(rr + (size_t)(m + 8) * 4096, 0, 1);
            __builtin_prefetch(ri + (size_t)(m + 8) * 4096, 0, 1);
        }
        float cr = tcs[2 * m], ci = tcs[2 * m + 1];
        float vr = rr[(size_t)m * 4096];
        float vi = ri[(size_t)m * 4096];
        ar = fmaf(cr, vr, fmaf(-ci, vi, ar));
        ai = fmaf(cr, vi, fmaf( ci, vr, ai));
    }
    RTr[(size_t)l * 4096 + oi] = ar;
    RTi[(size_t)l * 4096 + oi] = ai;
}

// ---------------------------------------------------------------------------
// Kernel 7: HLM[m][l][c2_o] = sign(m) * sum_i flm[m][l][i] * Rt[l][o][i]
// (inverse-side (-1)^m for m<0 folded here). grid (64, 127), block 64 (= o).
// ---------------------------------------------------------------------------
__global__ void k_applyR(const float* __restrict__ FLM,
                         const float* __restrict__ RTr,
                         const float* __restrict__ RTi,
                         float* __restrict__ HLM) {
    __shared__ float fs[C2];
    int l = blockIdx.x, m = blockIdx.y, o = threadIdx.x;
    const float* f = FLM + ((size_t)m * LF + l) * C2;
    fs[o]      = f[o];
    fs[NC + o] = f[NC + o];
    __syncthreads();
    const float* rr = RTr + ((size_t)l * NC + o) * NC;
    const float* ri = RTi + ((size_t)l * NC + o) * NC;
    float outre = 0.f, outim = 0.f;
    for (int i = 0; i < NC; ++i) {
        float fr = fs[i], fi = fs[NC + i];
        float vr = rr[i], vi = ri[i];
        outre = fmaf(fr, vr, fmaf(-fi, vi, outre));
        outim = fmaf(fr, vi, fmaf( fi, vr, outim));
    }
    int ms = m - 63;
    float sign = (ms < 0 && (ms & 1)) ? -1.f : 1.f;
    float* dst = HLM + ((size_t)m * LF + l) * C2;
    dst[o]      = outre * sign;
    dst[NC + o] = outim * sign;
}

// ---------------------------------------------------------------------------
// Kernel 8: inverse Legendre contraction via f32 WMMA, one block per m.
//   G[t][m][c2] = sum_l P[l,|m|,t] * HLM[m][l][c2]     (128 x 64 x 128 GEMM)
// ---------------------------------------------------------------------------
__global__ void k_gemm_inv(const float* __restrict__ P,
                           const float* __restrict__ HLM,
                           float* __restrict__ G) {
    int m = blockIdx.x;
    int mabs = (m < 63) ? (63 - m) : (m - 63);
    int lane  = threadIdx.x & 31;
    int wave  = threadIdx.x >> 5;
    int lhalf = lane >> 4;
    int l15   = lane & 15;
    int n0    = wave * 16;
    const float* Bm = HLM + (size_t)m * LF * C2;   // [l][c2]
    for (int r = 0; r < 8; ++r) {
        int t0 = r * 16;
        v8f acc = {};
        for (int k0 = 0; k0 < LF; k0 += 4) {
            int ka = k0 + lhalf * 2;       // l index
            v2f a, b;
            a.x = P[((size_t)ka * LF + mabs) * NT + t0 + l15];
            a.y = P[((size_t)(ka + 1) * LF + mabs) * NT + t0 + l15];
            b.x = Bm[(size_t)ka * C2 + n0 + l15];
            b.y = Bm[(size_t)(ka + 1) * C2 + n0 + l15];
            acc = __builtin_amdgcn_wmma_f32_16x16x4_f32(
                false, a, false, b, (short)0, acc, false, false);
        }
#pragma unroll
        for (int rv = 0; rv < 8; ++rv) {
            int trow = t0 + rv + lhalf * 8;
            G[((size_t)trow * MF + m) * C2 + n0 + l15] = acc[rv];
        }
    }
}

// ---------------------------------------------------------------------------
// Kernel 9: inverse phi-DFT.  out[t,p,c] = Re( sum_m G[t,m,c] e^{+2pi i p m/255} )
// One block per t; G[t] staged in LDS (65,024 B).
// ---------------------------------------------------------------------------
__global__ void k_idft(const float* __restrict__ G,
                       const float* __restrict__ twc,
                       const float* __restrict__ tws,
                       float* __restrict__ out) {
    __shared__ float gs[MF * C2];
    int t = blockIdx.x;
    int tid = threadIdx.x;
    const float* gt = G + (size_t)t * MF * C2;
    for (int i = tid; i < MF * C2; i += 256) gs[i] = gt[i];
    __syncthreads();
    for (int idx = tid; idx < NPHI * NC; idx += 256) {
        int p = idx >> 6;         // 0..254
        int c = idx & 63;
        // start at m = -63: k = ((255-63)*p) % 255, step +p per m
        int k = (int)(((long long)(NPHI - 63) * p) % NPHI);
        float acc = 0.f;
        for (int mi = 0; mi < MF; ++mi) {
            float gr = gs[mi * C2 + c];
            float gi = gs[mi * C2 + NC + c];
            acc = fmaf(gr, twc[k], fmaf(-gi, tws[k], acc));
            k += p; if (k >= NPHI) k -= NPHI;
        }
        out[((size_t)t * NPHI + p) * NC + c] = acc;
    }
}

// ---------------------------------------------------------------------------
extern "C" void kernel_launch(void* const* d_in, const int* in_sizes, int n_in,
                              void* d_out, int out_size, void* d_ws, size_t ws_size,
                              hipStream_t stream) {
    const float* x     = (const float*)d_in[0];   // (128, 255, 64)
    const float* temb  = (const float*)d_in[1];   // (256,)
    const float* Areal = (const float*)d_in[2];   // (64, 127, 256)
    const float* Aimag = (const float*)d_in[3];
    const float* Rreal = (const float*)d_in[4];   // (64, 127, 64, 64)
    const float* Rimag = (const float*)d_in[5];
    float* out = (float*)d_out;                   // (128, 255, 64)

    float* ws  = (float*)d_ws;
    float* TWC = ws;                                  // 256
    float* TWS = TWC + 256;                           // 256
    float* Q   = TWS + 256;                           // 128
    float* P   = Q   + 128;                           // 64*64*128   = 524288
    float* FW  = P   + 524288;                        // 127*128*128 = 2080768
    float* FLM = FW  + 2080768;                       // 127*64*128  = 1040384
    float* TC  = FLM + 1040384;                       // 127*64*2    = 16256
    float* RTr = TC  + 16256;                         // 64*64*64    = 262144
    float* RTi = RTr + 262144;                        // 262144
    float* HLM = RTi + 262144;                        // 1040384
    float* G   = HLM + 1040384;                       // 128*127*128 = 2080768

    k_precompute<<<1, 256, 0, stream>>>(TWC, TWS, Q);
    k_legendre <<<64, 128, 0, stream>>>(P);
    k_dft_fwd  <<<NT, 256, 0, stream>>>(x, TWC, TWS, Q, FW);
    k_gemm_fwd <<<MF, 256, 0, stream>>>(P, FW, FLM);
    k_modulate <<<MF, LF, 0, stream>>>(temb, Areal, Aimag, TC);
    k_mixR     <<<dim3(16, 64), 256, 0, stream>>>(TC, Rreal, Rimag, RTr, RTi);
    k_applyR   <<<dim3(LF, MF), NC, 0, stream>>>(FLM, RTr, RTi, HLM);
    k_gemm_inv <<<MF, 256, 0, stream>>>(P, HLM, G);
    k_idft     <<<NT, 256, 0, stream>>>(G, TWC, TWS, out);
}